// Entangle_53893249630764
// MI455X (gfx1250) — compile-verified
//
#include <hip/hip_runtime.h>
#include <math.h>

// ---------------------------------------------------------------------------
// Entangle kernel for gfx1250 (MI455X).
// Shapes: B=8, N=8, C=11, S=384, half=193. Rows (b,n,c) flat = 704.
// Pipeline (all f32, all GEMMs on v_wmma_f32_16x16x4_f32):
//   K0a fill DFT matrices Dre/Dim (384x384), K0b fill IRFFT matrix CIR (400x384)
//   K1  Sig = X @ D              (FFT as GEMM)
//   K2  per-(j,c,t) softmax stats over s of |km|   (LDS reduction)
//   K3a SM[jc] = Sig_rows(b,i;c) @ mask[jc]        (mask built on the fly)
//   K3b TM[jc] = Sig_rows(b;j,c) @ mask[jc]^T
//   K4  Z = cos(pol_i)*Sum_j mix*sig_j*SM + sin(pol_i)*sig_i*Sum_j mix*TM ; w
//   K5  y = (Zpk @ CIR + w*x) / 8                  (IRFFT as GEMM)
// ---------------------------------------------------------------------------

typedef __attribute__((ext_vector_type(2))) float v2f;
typedef __attribute__((ext_vector_type(8))) float v8f;

#define S_LEN   384
#define HALF    193
#define NB      8
#define NN      8
#define NC      11
#define NROWS   704      // B*N*C
#define NJC     88       // N*C
#define SMW     208      // padded half (13*16)
#define ZW      400      // packed Z width: [0,193) re, [200,393) im, rest zero
#define TWO_PI_OVER_S (6.283185307179586476925f / 384.0f)

__device__ __forceinline__ v8f wmma4(v2f a, v2f b, v8f c) {
  // D(16x16) = A(16x4) * B(4x16) + C  (f32)
  return __builtin_amdgcn_wmma_f32_16x16x4_f32(false, a, false, b, (short)0, c,
                                               false, false);
}

// ---------------- K0a: DFT matrices  D[s][f] = e^{-2*pi*i*s*f/S} -------------
__global__ void fill_dft(float* __restrict__ Dre, float* __restrict__ Dim) {
  int idx = blockIdx.x * blockDim.x + threadIdx.x;   // 384*384 exact
  int s = idx / S_LEN, f = idx % S_LEN;
  int p = (s * f) % S_LEN;                           // exact integer phase
  float ang = (float)p * TWO_PI_OVER_S;
  Dre[idx] = cosf(ang);
  Dim[idx] = -sinf(ang);
}

// ---------------- K0b: IRFFT matrix (numpy irfft semantics) ------------------
// y[t] = (1/S)[ Re(X0) + 2*sum_{1..191} Re(Xf e^{i*2pi*f*t/S}) + Re(X192)*cos(pi t) ]
__global__ void fill_cir(float* __restrict__ CIR) {
  int idx = blockIdx.x * blockDim.x + threadIdx.x;   // 400*384 exact
  int k = idx / S_LEN, t = idx % S_LEN;
  float val = 0.0f;
  if (k < HALF || (k >= 200 && k < 200 + HALF)) {
    bool isim = (k >= 200);
    int f = isim ? (k - 200) : k;
    float w = (f == 0 || f == 192) ? 1.0f : 2.0f;
    int p = (f * t) % S_LEN;
    float ang = (float)p * TWO_PI_OVER_S;
    val = isim ? (-(w / 384.0f) * sinf(ang)) : ((w / 384.0f) * cosf(ang));
  }
  CIR[idx] = val;
}

// ---------------- K1: Sig = X @ D (two outputs share A loads) ----------------
__global__ void fft_gemm(const float* __restrict__ X, const float* __restrict__ Dre,
                         const float* __restrict__ Dim, float* __restrict__ SigRe,
                         float* __restrict__ SigIm) {
  int wave = blockIdx.x * (blockDim.x >> 5) + (threadIdx.x >> 5);
  if (wave >= 44 * 24) return;                       // wave-uniform exit
  int mt = wave / 24, nt = wave % 24;
  int lane = threadIdx.x & 31, g = lane >> 4, hl = lane & 15;
  int arow = mt * 16 + hl;
  int ncol = nt * 16 + hl;
  v8f accR = {}, accI = {};
  for (int k = 0; k < S_LEN; k += 4) {
    int ka = k + 2 * g;
    v2f a;  a.x  = X[arow * S_LEN + ka];        a.y  = X[arow * S_LEN + ka + 1];
    v2f br; br.x = Dre[ka * S_LEN + ncol];      br.y = Dre[(ka + 1) * S_LEN + ncol];
    v2f bi; bi.x = Dim[ka * S_LEN + ncol];      bi.y = Dim[(ka + 1) * S_LEN + ncol];
    accR = wmma4(a, br, accR);
    accI = wmma4(a, bi, accI);
  }
  for (int v = 0; v < 8; ++v) {
    int m = mt * 16 + v + 8 * g;
    SigRe[m * S_LEN + ncol] = accR[v];
    SigIm[m * S_LEN + ncol] = accI[v];
  }
}

// ---------------- K2: softmax stats over s for each (j,c,t) ------------------
__global__ void softmax_stats(const float* __restrict__ km, float* __restrict__ stM,
                              float* __restrict__ stS) {
  int tb = blockIdx.x % 12;            // t block of 32
  int jc = blockIdx.x / 12;
  int tx = threadIdx.x & 31;           // t within block
  int ty = threadIdx.x >> 5;           // s group (8 waves)
  int t = tb * 32 + tx;
  const float* kmB = km + (size_t)jc * S_LEN * S_LEN * 2;
  float m = -1e30f, ssum = 0.0f;
  for (int s = ty; s < S_LEN; s += 8) {                  // online softmax
    size_t e = ((size_t)s * S_LEN + t) * 2;
    float re = kmB[e], im = kmB[e + 1];
    float a = sqrtf(re * re + im * im);
    float nm = fmaxf(m, a);
    ssum = ssum * __expf(m - nm) + __expf(a - nm);
    m = nm;
  }
  __shared__ float shM[8][32];
  __shared__ float shS[8][32];
  shM[ty][tx] = m; shS[ty][tx] = ssum;
  __syncthreads();
  if (ty == 0) {
    float M = shM[0][tx], Sv = shS[0][tx];
    for (int q = 1; q < 8; ++q) {
      float m2 = shM[q][tx], s2 = shS[q][tx];
      float NM = fmaxf(M, m2);
      Sv = Sv * __expf(M - NM) + s2 * __expf(m2 - NM);
      M = NM;
    }
    stM[jc * S_LEN + t] = M;
    stS[jc * S_LEN + t] = Sv;
  }
}

// mask element from km + stats:  km * softmax(|km|)/|km|  (angle preserved)
__device__ __forceinline__ void mask_elem(float re, float im, float mx,
                                          float inv_den, float& mre, float& mim) {
  float a = sqrtf(re * re + im * im);
  float p = __expf(a - mx) * inv_den;
  if (a > 0.0f) { float sc = p / a; mre = re * sc; mim = im * sc; }
  else          { mre = p; mim = 0.0f; }             // angle(0)=0 -> e^{i0}=1
}

// ---------------- K3a: SM[jc][bi][t] = sum_s Sig[bi*11+c][s] * mask[jc][s][t]
__global__ void sm_gemm(const float* __restrict__ SigRe, const float* __restrict__ SigIm,
                        const float* __restrict__ km, const float* __restrict__ stM,
                        const float* __restrict__ stS, float* __restrict__ SMre,
                        float* __restrict__ SMim) {
  int wave = blockIdx.x * (blockDim.x >> 5) + (threadIdx.x >> 5);
  if (wave >= NJC * 4 * 13) return;
  int nt = wave % 13; int tmp = wave / 13;
  int mt = tmp & 3;   int jc = tmp >> 2;
  int lane = threadIdx.x & 31, g = lane >> 4, hl = lane & 15;
  int c = jc % NC;
  int t = nt * 16 + hl;                              // 0..207 (<384, always valid read)
  float mx = stM[jc * S_LEN + t];
  float inv_den = 1.0f / stS[jc * S_LEN + t];
  int bi = mt * 16 + hl;
  int arow = bi * NC + c;                            // Sig row (b*8+i)*11+c
  const float* kmB = km + (size_t)jc * S_LEN * S_LEN * 2;
  v8f P = {}, Q = {}, R = {}, T = {};                // f32 WMMA has no A/B negate
  for (int k = 0; k < S_LEN; k += 4) {
    int s0 = k + 2 * g;
    v2f are, aim;
    are.x = SigRe[arow * S_LEN + s0]; are.y = SigRe[arow * S_LEN + s0 + 1];
    aim.x = SigIm[arow * S_LEN + s0]; aim.y = SigIm[arow * S_LEN + s0 + 1];
    size_t e0 = ((size_t)s0 * S_LEN + t) * 2;
    size_t e1 = ((size_t)(s0 + 1) * S_LEN + t) * 2;
    __builtin_prefetch((const void*)(kmB + e0 + (size_t)4 * S_LEN * 2), 0, 0);
    v2f bre, bim;
    { float r0, i0; mask_elem(kmB[e0], kmB[e0 + 1], mx, inv_den, r0, i0);
      bre.x = r0; bim.x = i0; }
    { float r1, i1; mask_elem(kmB[e1], kmB[e1 + 1], mx, inv_den, r1, i1);
      bre.y = r1; bim.y = i1; }
    P = wmma4(are, bre, P);   // Re*Re
    Q = wmma4(aim, bim, Q);   // Im*Im
    R = wmma4(are, bim, R);   // Re*Im
    T = wmma4(aim, bre, T);   // Im*Re
  }
  for (int v = 0; v < 8; ++v) {
    int m = mt * 16 + v + 8 * g;
    size_t o = ((size_t)jc * 64 + m) * SMW + (nt * 16 + hl);
    SMre[o] = P[v] - Q[v];
    SMim[o] = R[v] + T[v];
  }
}

// ---------------- K3b: TM[jc][b][s] = sum_t Sig[(b*8+j)*11+c][t] * mask[jc][s][t]
__global__ void tm_gemm(const float* __restrict__ SigRe, const float* __restrict__ SigIm,
                        const float* __restrict__ km, const float* __restrict__ stM,
                        const float* __restrict__ stS, float* __restrict__ TMre,
                        float* __restrict__ TMim) {
  int wave = blockIdx.x * (blockDim.x >> 5) + (threadIdx.x >> 5);
  if (wave >= NJC * 13) return;
  int nt = wave % 13; int jc = wave / 13;
  int lane = threadIdx.x & 31, g = lane >> 4, hl = lane & 15;
  int c = jc % NC, j = jc / NC;
  bool vrow = (hl < NB);                             // only 8 real rows, pad to 16
  int arow = (((hl & 7) * NN + j) * NC + c);         // Sig row (b*8+j)*11+c
  int scol = nt * 16 + hl;                           // s index (0..207 < 384)
  const float* kmB = km + (size_t)jc * S_LEN * S_LEN * 2;
  v8f P = {}, Q = {}, R = {}, T = {};
  for (int k = 0; k < S_LEN; k += 4) {
    int t0 = k + 2 * g;
    v2f are, aim;
    are.x = vrow ? SigRe[arow * S_LEN + t0]     : 0.0f;
    are.y = vrow ? SigRe[arow * S_LEN + t0 + 1] : 0.0f;
    aim.x = vrow ? SigIm[arow * S_LEN + t0]     : 0.0f;
    aim.y = vrow ? SigIm[arow * S_LEN + t0 + 1] : 0.0f;
    float mx0 = stM[jc * S_LEN + t0],     id0 = 1.0f / stS[jc * S_LEN + t0];
    float mx1 = stM[jc * S_LEN + t0 + 1], id1 = 1.0f / stS[jc * S_LEN + t0 + 1];
    size_t e0 = ((size_t)scol * S_LEN + t0) * 2;     // transposed km read
    v2f bre, bim;
    { float r0, i0; mask_elem(kmB[e0],     kmB[e0 + 1], mx0, id0, r0, i0);
      bre.x = r0; bim.x = i0; }
    { float r1, i1; mask_elem(kmB[e0 + 2], kmB[e0 + 3], mx1, id1, r1, i1);
      bre.y = r1; bim.y = i1; }
    P = wmma4(are, bre, P);
    Q = wmma4(aim, bim, Q);
    R = wmma4(are, bim, R);
    T = wmma4(aim, bre, T);
  }
  for (int v = 0; v < 8; ++v) {
    int m = v + 8 * g;                               // padded b row (0..15)
    size_t o = ((size_t)jc * 16 + m) * SMW + scol;
    TMre[o] = P[v] - Q[v];
    TMim[o] = R[v] + T[v];
  }
}

// ---------------- K4: j-reduction with mix; pack Z; emit w -------------------
__global__ void combine_k(const float* __restrict__ SigRe, const float* __restrict__ SigIm,
                          const float* __restrict__ SMre, const float* __restrict__ SMim,
                          const float* __restrict__ TMre, const float* __restrict__ TMim,
                          const float* __restrict__ pol, const float* __restrict__ gmean,
                          const float* __restrict__ gstd, float* __restrict__ Zpk,
                          float* __restrict__ wbuf) {
  int idx = blockIdx.x * blockDim.x + threadIdx.x;   // 704*200 exact
  int f = idx % 200;
  int r = idx / 200;                                 // (b*8+i)*11+c
  int c = r % NC, bi = r / NC;
  int i = bi & 7, b = bi >> 3;
  if (f >= HALF) {                                   // zero the pack padding
    Zpk[r * ZW + f] = 0.0f;
    Zpk[r * ZW + 200 + f] = 0.0f;
    return;
  }
  float dci = SigRe[r * S_LEN];                      // DC bin = sum_s x (real)
  float mean = gmean[i];
  float istd = 1.0f / gstd[i];
  float Ar = 0.f, Ai = 0.f, Tr = 0.f, Ti = 0.f, msum = 0.f;
  for (int j = 0; j < NN; ++j) {
    int rj = (b * NN + j) * NC + c;
    float dcj = SigRe[rj * S_LEN];
    // corr = mean_t irfft(prod)[t] = Re(prod[0])/S = dci*dcj/S
    float z = (dci * dcj * (1.0f / 384.0f) - mean) * istd;
    float mix = __expf(-0.5f * z * z);
    msum += mix;
    int jc = j * NC + c;
    float sjr = SigRe[rj * S_LEN + f], sji = SigIm[rj * S_LEN + f];
    size_t os = ((size_t)jc * 64 + bi) * SMW + f;
    float smr = SMre[os], smi = SMim[os];
    Ar += mix * (sjr * smr - sji * smi);             // colA = sig_j * sm
    Ai += mix * (sjr * smi + sji * smr);
    size_t ot = ((size_t)jc * 16 + b) * SMW + f;
    Tr += mix * TMre[ot];
    Ti += mix * TMim[ot];
  }
  float sir = SigRe[r * S_LEN + f], sii = SigIm[r * S_LEN + f];
  float Br = sir * Tr - sii * Ti;                    // colB = sig_i * (sum mix*tm)
  float Bi = sir * Ti + sii * Tr;
  float cp = cosf(pol[i]), sp = sinf(pol[i]);
  Zpk[r * ZW + f]       = cp * Ar + sp * Br;
  Zpk[r * ZW + 200 + f] = cp * Ai + sp * Bi;
  if (f == 0) wbuf[r] = (float)NN - msum;            // sum_j (1 - mix)
}

// ---------------- K5: y = (Zpk @ CIR + w*x) / N ------------------------------
__global__ void irfft_gemm(const float* __restrict__ Zpk, const float* __restrict__ CIR,
                           const float* __restrict__ X, const float* __restrict__ wbuf,
                           float* __restrict__ Y) {
  int wave = blockIdx.x * (blockDim.x >> 5) + (threadIdx.x >> 5);
  if (wave >= 44 * 24) return;
  int mt = wave / 24, nt = wave % 24;
  int lane = threadIdx.x & 31, g = lane >> 4, hl = lane & 15;
  int arow = mt * 16 + hl;
  int ncol = nt * 16 + hl;
  v8f acc = {};
  for (int k = 0; k < ZW; k += 4) {
    int kk = k + 2 * g;
    v2f a;  a.x = Zpk[arow * ZW + kk];      a.y = Zpk[arow * ZW + kk + 1];
    v2f bb; bb.x = CIR[kk * S_LEN + ncol];  bb.y = CIR[(kk + 1) * S_LEN + ncol];
    acc = wmma4(a, bb, acc);
  }
  for (int v = 0; v < 8; ++v) {
    int m = mt * 16 + v + 8 * g;
    Y[m * S_LEN + ncol] = (acc[v] + wbuf[m] * X[m * S_LEN + ncol]) * 0.125f;
  }
}

// ---------------------------------------------------------------------------
extern "C" void kernel_launch(void* const* d_in, const int* in_sizes, int n_in,
                              void* d_out, int out_size, void* d_ws, size_t ws_size,
                              hipStream_t stream) {
  (void)in_sizes; (void)n_in; (void)out_size; (void)ws_size;
  const float* x   = (const float*)d_in[0];   // (8,8,11,384) f32
  const float* km  = (const float*)d_in[1];   // (8,11,384,384) c64 -> interleaved f32
  const float* pol = (const float*)d_in[2];   // (8,)
  const float* gm  = (const float*)d_in[3];   // (8,)
  const float* gs  = (const float*)d_in[4];   // (8,)
  float* y  = (float*)d_out;                  // (8,8,11,384) f32
  float* ws = (float*)d_ws;                   // ~17.1 MB used

  float* Dre   = ws;                    // 384*384
  float* Dim   = Dre   + 147456;        // 384*384
  float* CIR   = Dim   + 147456;        // 400*384
  float* SigRe = CIR   + 153600;        // 704*384
  float* SigIm = SigRe + 270336;        // 704*384
  float* stM   = SigIm + 270336;        // 88*384
  float* stS   = stM   + 33792;         // 88*384
  float* SMre  = stS   + 33792;         // 88*64*208
  float* SMim  = SMre  + 1171456;
  float* TMre  = SMim  + 1171456;       // 88*16*208
  float* TMim  = TMre  + 292864;
  float* Zpk   = TMim  + 292864;        // 704*400
  float* wbuf  = Zpk   + 281600;        // 704

  fill_dft     <<<576,  256, 0, stream>>>(Dre, Dim);
  fill_cir     <<<600,  256, 0, stream>>>(CIR);
  fft_gemm     <<<264,  128, 0, stream>>>(x, Dre, Dim, SigRe, SigIm);
  softmax_stats<<<1056, 256, 0, stream>>>(km, stM, stS);
  sm_gemm      <<<1144, 128, 0, stream>>>(SigRe, SigIm, km, stM, stS, SMre, SMim);
  tm_gemm      <<<286,  128, 0, stream>>>(SigRe, SigIm, km, stM, stS, TMre, TMim);
  combine_k    <<<550,  256, 0, stream>>>(SigRe, SigIm, SMre, SMim, TMre, TMim,
                                          pol, gm, gs, Zpk, wbuf);
  irfft_gemm   <<<264,  128, 0, stream>>>(Zpk, CIR, x, wbuf, y);
}